// PriorBNNmodel_60902636257840
// MI455X (gfx1250) — compile-verified
//
#include <hip/hip_runtime.h>
#include <math.h>
#include <stdint.h>

// ---------------------------------------------------------------------------
// Problem constants (from the reference)
// ---------------------------------------------------------------------------
#define TT     1024
#define NPIX   1024
#define PTOT   14723          // sum of PARAM_SIZES (odd -> w_vec rows only 4B aligned)
#define CUM1   6208
#define CUM2   10368
#define CUM3   14528
#define W0F    30.0f

typedef __attribute__((ext_vector_type(2))) float v2f;
typedef __attribute__((ext_vector_type(8))) float v8f;

// V_WMMA_F32_16X16X4_F32:  D(16x16,f32) = A(16x4,f32) x B(4x16,f32) + C
__device__ __forceinline__ v8f wmma4(v2f a, v2f b, v8f c) {
    return __builtin_amdgcn_wmma_f32_16x16x4_f32(
        false, a, false, b, (short)0, c, false, false);
}

__device__ __forceinline__ float st_f(float x) {
    float sp = (x > 20.0f) ? x : log1pf(__expf(x));
    return sp * (1.0f / 6.0f);
}

// ---------------------------------------------------------------------------
// K1: w_vec = loc + st(log_scale) * eps_w
// ---------------------------------------------------------------------------
__global__ __launch_bounds__(256) void k_wvec(
    const float* __restrict__ loc, const float* __restrict__ lsc,
    const float* __restrict__ eps, float* __restrict__ wv, int n) {
    int i = blockIdx.x * 256 + threadIdx.x;
    if (i < n) wv[i] = loc[i] + st_f(lsc[i]) * eps[i];
}

// ---------------------------------------------------------------------------
// K2a: main GEMM, NO guards (requires p%64==0).  g_slice = wv_slice @ A.
// Block = 256 thr = 8 waves; block tile 128(M) x 64(N); K-step 16.
// B tile (16x64) staged in LDS via async global->LDS b128, double-buffered;
// all 8 waves share it.  Each wave: 16x64 D-tile, 4 f32 WMMA accumulators.
// ---------------------------------------------------------------------------
__global__ __launch_bounds__(256) void k_gemm_main(
    const float* __restrict__ wv, const float* __restrict__ A,
    float* __restrict__ g, int p, int cum) {
    __shared__ __align__(16) float Bs[2][16 * 64];   // double-buffered B tile

    const int tid  = threadIdx.x;
    const int lane = tid & 31;
    const int wave = tid >> 5;
    const int hl   = lane >> 4;        // half-wave 0|1
    const int ll   = lane & 15;
    const int m0   = blockIdx.x * 128 + wave * 16;   // t-row tile
    const int n0   = blockIdx.y * 64;                // col tile

    // this thread's slice of the B-tile staging (one b128 per K-step)
    const int brow = tid >> 4;            // 0..15  (K row within tile)
    const int bcol = (tid & 15) * 4;      // 0..60  (N col, 4 floats)
    const unsigned lds_b0 = (unsigned)(uintptr_t)&Bs[0][brow * 64 + bcol];
    const unsigned lds_b1 = (unsigned)(uintptr_t)&Bs[1][brow * 64 + bcol];

    v8f acc0 = {}, acc1 = {}, acc2 = {}, acc3 = {};
    const float* wrow = wv + (size_t)(m0 + ll) * PTOT + cum;

    const int nks = p >> 4;   // K-steps of 16

    // prologue: issue async stage of tile 0 into buffer 0
    {
        uint64_t src = (uint64_t)(uintptr_t)(A + (size_t)brow * p + (n0 + bcol));
        asm volatile("global_load_async_to_lds_b128 %0, %1, off"
                     :: "v"(lds_b0), "v"(src) : "memory");
    }

    for (int ks = 0; ks < nks; ++ks) {
        const int cur = ks & 1;
        const bool more = (ks + 1) < nks;
        if (more) {   // prefetch next tile into the other buffer (2 in flight)
            uint64_t src = (uint64_t)(uintptr_t)
                (A + (size_t)((ks + 1) * 16 + brow) * p + (n0 + bcol));
            asm volatile("global_load_async_to_lds_b128 %0, %1, off"
                         :: "v"(cur ? lds_b0 : lds_b1), "v"(src) : "memory");
            asm volatile("s_wait_asynccnt 0x1" ::: "memory");
        } else {
            asm volatile("s_wait_asynccnt 0x0" ::: "memory");
        }
        __syncthreads();                       // current B tile visible

        const float* Bb = &Bs[cur][0];
        const int kb = ks * 16;
        #pragma unroll
        for (int kk = 0; kk < 16; kk += 4) {
            const int ka = kk + 2 * hl;        // this half-wave's K pair
            v2f a;                             // A frag: M=ll, K=kb+ka, kb+ka+1
            a.x = wrow[kb + ka];
            a.y = wrow[kb + ka + 1];
            v2f b0 = { Bb[ka * 64 + ll],      Bb[ka * 64 + 64 + ll]      };
            v2f b1 = { Bb[ka * 64 + 16 + ll], Bb[ka * 64 + 64 + 16 + ll] };
            v2f b2 = { Bb[ka * 64 + 32 + ll], Bb[ka * 64 + 64 + 32 + ll] };
            v2f b3 = { Bb[ka * 64 + 48 + ll], Bb[ka * 64 + 64 + 48 + ll] };
            acc0 = wmma4(a, b0, acc0);
            acc1 = wmma4(a, b1, acc1);
            acc2 = wmma4(a, b2, acc2);
            acc3 = wmma4(a, b3, acc3);
        }
        __syncthreads();                       // all reads done before overwrite
    }

    // Store: lane holds N=ll, rows r + 8*hl
    #pragma unroll
    for (int nt = 0; nt < 4; ++nt) {
        const int n = n0 + nt * 16 + ll;
        const v8f acc = (nt == 0) ? acc0 : (nt == 1) ? acc1 : (nt == 2) ? acc2 : acc3;
        #pragma unroll
        for (int r = 0; r < 8; ++r)
            g[(size_t)(m0 + r + 8 * hl) * PTOT + cum + n] = acc[r];
    }
}

// ---------------------------------------------------------------------------
// K2b: guarded edge GEMM (layer 3 only, p=195 -- tiny)
// ---------------------------------------------------------------------------
__global__ __launch_bounds__(256) void k_gemm_edge(
    const float* __restrict__ wv, const float* __restrict__ A,
    float* __restrict__ g, int p, int cum) {
    const int lane = threadIdx.x & 31;
    const int wave = threadIdx.x >> 5;
    const int hl = lane >> 4, ll = lane & 15;
    const int m0 = blockIdx.x * 128 + wave * 16;
    const int n0 = blockIdx.y * 64;

    v8f acc0 = {}, acc1 = {}, acc2 = {}, acc3 = {};
    const float* wrow = wv + (size_t)(m0 + ll) * PTOT + cum;

    for (int k = 0; k < p; k += 4) {
        const int ka = k + 2 * hl;
        v2f a;
        a.x = (ka     < p) ? wrow[ka]     : 0.0f;
        a.y = (ka + 1 < p) ? wrow[ka + 1] : 0.0f;
        const size_t r0 = (size_t)ka * p, r1 = (size_t)(ka + 1) * p;
        const bool k0 = (ka < p), k1 = (ka + 1 < p);
        v2f b[4];
        #pragma unroll
        for (int nt = 0; nt < 4; ++nt) {
            int n = n0 + nt * 16 + ll;
            bool nok = (n < p);
            b[nt].x = (nok && k0) ? A[r0 + n] : 0.0f;
            b[nt].y = (nok && k1) ? A[r1 + n] : 0.0f;
        }
        acc0 = wmma4(a, b[0], acc0);
        acc1 = wmma4(a, b[1], acc1);
        acc2 = wmma4(a, b[2], acc2);
        acc3 = wmma4(a, b[3], acc3);
    }
    #pragma unroll
    for (int nt = 0; nt < 4; ++nt) {
        const int n = n0 + nt * 16 + ll;
        if (n >= p) continue;
        const v8f acc = (nt == 0) ? acc0 : (nt == 1) ? acc1 : (nt == 2) ? acc2 : acc3;
        #pragma unroll
        for (int r = 0; r < 8; ++r)
            g[(size_t)(m0 + r + 8 * hl) * PTOT + cum + n] = acc[r];
    }
}

// ---------------------------------------------------------------------------
// K3: fused 4-layer per-t MLP.  grid = (T), block = 256 (8 waves).
// Weights staged ONCE per t, reused for all 8 pixel tiles (1024 pixels).
// ---------------------------------------------------------------------------
template <int DIN, int NT, bool ACT>
__device__ __forceinline__ void mlp_layer(
    float* hw, const float* __restrict__ Ws, const float* __restrict__ bs,
    int lane) {
    const int hl = lane >> 4, ll = lane & 15;
    v8f acc[NT];
    #pragma unroll
    for (int i = 0; i < NT; ++i) acc[i] = (v8f){0, 0, 0, 0, 0, 0, 0, 0};

    #pragma unroll
    for (int k = 0; k < DIN; k += 4) {
        const int ka = k + 2 * hl;
        v2f a = *(const v2f*)&hw[ll * 96 + ka];
        #pragma unroll
        for (int nt = 0; nt < NT; ++nt) {
            const int n = nt * 16 + ll;
            v2f b;
            b.x = Ws[(size_t)ka * (NT * 16) + n];
            b.y = Ws[(size_t)(ka + 1) * (NT * 16) + n];
            acc[nt] = wmma4(a, b, acc[nt]);
        }
    }
    #pragma unroll
    for (int nt = 0; nt < NT; ++nt) {
        const int n = nt * 16 + ll;
        const float bias = bs[n];
        #pragma unroll
        for (int r = 0; r < 8; ++r) {
            float v = acc[nt][r] + bias;
            if (ACT) v = __sinf(W0F * v);
            hw[(r + 8 * hl) * 96 + n] = v;
        }
    }
}

__global__ __launch_bounds__(256) void k_mlp(
    const float* __restrict__ x, const float* __restrict__ lpe_loc,
    const float* __restrict__ lpe_ls, const float* __restrict__ eps_lpe,
    const float* __restrict__ g, float* __restrict__ out) {
    __shared__ __align__(16) float W0s[96 * 64];
    __shared__ __align__(16) float W1s[64 * 64];
    __shared__ __align__(16) float W2s[64 * 64];
    __shared__ __align__(16) float W3s[64 * 16];      // 64x3 padded to 64x16
    __shared__ float b0s[64], b1s[64], b2s[64], b3s[16];
    __shared__ __align__(16) float hbuf[8 * 16 * 96]; // per-wave h staging

    const int t = blockIdx.x;
    const int tid = threadIdx.x;
    const int wave = tid >> 5;
    const int lane = tid & 31;
    const float* gt = g + (size_t)t * PTOT;

    // --- stage per-t weights/biases once ---
    if (tid < 64) { b0s[tid] = gt[tid];
                    b1s[tid] = gt[CUM1 + tid];
                    b2s[tid] = gt[CUM2 + tid]; }
    if (tid < 16)   b3s[tid] = (tid < 3) ? gt[CUM3 + tid] : 0.0f;
    for (int i = tid; i < 96 * 64; i += 256) W0s[i] = gt[64 + i];
    for (int i = tid; i < 64 * 64; i += 256) {
        W1s[i] = gt[CUM1 + 64 + i];
        W2s[i] = gt[CUM2 + 64 + i];
    }
    for (int i = tid; i < 64 * 16; i += 256) {
        const int r = i >> 4, c = i & 15;
        W3s[i] = (c < 3) ? gt[CUM3 + 3 + r * 3 + c] : 0.0f;
    }
    __syncthreads();   // weights visible; hbuf is per-wave below (no more barriers)

    float* hw = hbuf + wave * 16 * 96;

    for (int tile = 0; tile < NPIX / 128; ++tile) {
        const int rowBase = tile * 128 + wave * 16;

        // build h0 = concat(x, pe) for this wave's 16-pixel tile
        for (int i = lane; i < 16 * 96; i += 32) {
            const int r = i / 96, c = i % 96;
            const size_t pix = (size_t)rowBase + r;
            float v;
            if (c < 64) {
                v = x[((size_t)t * NPIX + pix) * 64 + c];
            } else {
                const size_t idx = ((size_t)t * NPIX + pix) * 32 + (c - 64);
                v = lpe_loc[idx] + st_f(lpe_ls[idx]) * eps_lpe[idx];
            }
            hw[r * 96 + c] = v;
        }

        mlp_layer<96, 4, true>(hw, W0s, b0s, lane);
        mlp_layer<64, 4, true>(hw, W1s, b1s, lane);
        mlp_layer<64, 4, true>(hw, W2s, b2s, lane);

        // final 64 -> 3 (padded to 16), store
        {
            const int hl = lane >> 4, ll = lane & 15;
            v8f acc = {};
            #pragma unroll
            for (int k = 0; k < 64; k += 4) {
                const int ka = k + 2 * hl;
                v2f a = *(const v2f*)&hw[ll * 96 + ka];
                v2f b;
                b.x = W3s[ka * 16 + ll];
                b.y = W3s[(ka + 1) * 16 + ll];
                acc = wmma4(a, b, acc);
            }
            if (ll < 3) {
                const float bias = b3s[ll];
                #pragma unroll
                for (int r = 0; r < 8; ++r) {
                    const size_t row = (size_t)rowBase + r + 8 * hl;
                    out[((size_t)t * NPIX + row) * 3 + ll] = acc[r] + bias;
                }
            }
        }
    }
}

// ---------------------------------------------------------------------------
// Launcher
// ---------------------------------------------------------------------------
extern "C" void kernel_launch(void* const* d_in, const int* in_sizes, int n_in,
                              void* d_out, int out_size, void* d_ws, size_t ws_size,
                              hipStream_t stream) {
    (void)in_sizes; (void)n_in; (void)out_size; (void)ws_size;
    const float* x        = (const float*)d_in[0];
    const float* loc      = (const float*)d_in[1];
    const float* log_sc   = (const float*)d_in[2];
    const float* lpe_loc  = (const float*)d_in[3];
    const float* lpe_lsc  = (const float*)d_in[4];
    const float* eps_w    = (const float*)d_in[5];
    const float* eps_lpe  = (const float*)d_in[6];
    const float* A0       = (const float*)d_in[7];
    const float* A1       = (const float*)d_in[8];
    const float* A2       = (const float*)d_in[9];
    const float* A3       = (const float*)d_in[10];
    float* out = (float*)d_out;

    float* wv = (float*)d_ws;                       // T x PTOT
    float* g  = wv + (size_t)TT * PTOT;             // T x PTOT

    const int n = TT * PTOT;
    k_wvec<<<(n + 255) / 256, 256, 0, stream>>>(loc, log_sc, eps_w, wv, n);

    // layers 0-2: branch-free async-staged GEMM (p % 64 == 0)
    k_gemm_main<<<dim3(TT / 128, 6208 / 64), 256, 0, stream>>>(wv, A0, g, 6208, 0);
    k_gemm_main<<<dim3(TT / 128, 4160 / 64), 256, 0, stream>>>(wv, A1, g, 4160, CUM1);
    k_gemm_main<<<dim3(TT / 128, 4160 / 64), 256, 0, stream>>>(wv, A2, g, 4160, CUM2);
    // layer 3: tiny, guarded
    k_gemm_edge<<<dim3(TT / 128, (195 + 63) / 64), 256, 0, stream>>>(wv, A3, g, 195, CUM3);

    // fused per-t MLP (weights staged once per t, 8 pixel tiles)
    k_mlp<<<dim3(TT), 256, 0, stream>>>(x, lpe_loc, lpe_lsc, eps_lpe, g, out);
}